// MinibatchDiscrimination_15221364097123
// MI455X (gfx1250) — compile-verified
//
#include <hip/hip_runtime.h>
#include <hip/hip_bf16.h>

typedef __attribute__((ext_vector_type(2))) float v2f;
typedef __attribute__((ext_vector_type(8))) float v8f;

#define BATCH 1024
#define INF   512
#define JDIM  64
#define KDIM  8
#define JK    512   // JDIM * KDIM
#define OUTC  576   // INF + JDIM
#define CLAMP_MAX 10.0f
#define NEG_LOG2E -1.4426950408889634f

// ---------------------------------------------------------------------------
// Kernel 1: M[b, j*8+k] = sum_i x[b,i] * T[i, j*8+k]
// One wave computes a 16x16 tile of M via V_WMMA_F32_16X16X4_F32, K=512.
// A layout (16x4 f32): VGPR0 = {K=0 | K=2}, VGPR1 = {K=1 | K=3} across lane halves.
// C/D layout (16x16 f32): VGPR r -> M = r (lanes 0-15) / M = r+8 (lanes 16-31), N = lane&15.
// ---------------------------------------------------------------------------
__global__ __launch_bounds__(256) void gemm_m(const float* __restrict__ x,
                                              const float* __restrict__ T,
                                              float* __restrict__ M) {
    const int gtid = blockIdx.x * blockDim.x + threadIdx.x;
    const int wave = gtid >> 5;
    const int lane = threadIdx.x & 31;
    const int tilesPerRow = JK / 16;              // 32 column tiles
    const int tr = wave / tilesPerRow;            // 0..63
    const int tc = wave % tilesPerRow;            // 0..31
    const int row0 = tr * 16;
    const int col0 = tc * 16;
    const int m  = lane & 15;
    const int hi = lane >> 4;

    v8f acc = {};
    const float* xrow = x + (row0 + m) * INF;
    for (int k = 0; k < INF; k += 4) {
        // A: x rows, K-slice [k, k+4)
        v2f a = *(const v2f*)(xrow + k + 2 * hi);          // 8B-aligned (even offset)
        // B: T rows k+2hi and k+2hi+1, column col0+m
        v2f b;
        b.x = T[(k + 2 * hi    ) * JK + col0 + m];
        b.y = T[(k + 2 * hi + 1) * JK + col0 + m];
        acc = __builtin_amdgcn_wmma_f32_16x16x4_f32(false, a, false, b,
                                                    (short)0, acc, false, false);
    }
    #pragma unroll
    for (int r = 0; r < 8; ++r)
        M[(row0 + r + 8 * hi) * JK + col0 + m] = acc[r];
}

// ---------------------------------------------------------------------------
// Kernel 2: n2[b,j] = sum_k M[b,j,k]^2   (one thread per (b,j), 8 contiguous f32)
// ---------------------------------------------------------------------------
__global__ __launch_bounds__(256) void row_norms(const float* __restrict__ M,
                                                 float* __restrict__ n2) {
    const int idx = blockIdx.x * blockDim.x + threadIdx.x;
    if (idx < BATCH * JDIM) {
        const float4* p = (const float4*)(M + idx * KDIM);  // 32B aligned
        float4 u = p[0], v = p[1];
        n2[idx] = u.x * u.x + u.y * u.y + u.z * u.z + u.w * u.w +
                  v.x * v.x + v.y * v.y + v.z * v.z + v.w * v.w;
    }
}

// ---------------------------------------------------------------------------
// Kernel 3: feats[a,j] = sum_b exp(-min(||M[b,j,:] - M[a,j,:]||, 10))
// expanded as n2[a]+n2[b]-2*dot, with dot computed by WMMA (two K=4 steps).
// One wave per (a-tile of 16, j). Writes directly into out[:, 512+j].
// ---------------------------------------------------------------------------
__global__ __launch_bounds__(256) void pairwise_feats(const float* __restrict__ M,
                                                      const float* __restrict__ n2,
                                                      float* __restrict__ out) {
    const int gtid = blockIdx.x * blockDim.x + threadIdx.x;
    const int wave = gtid >> 5;
    const int lane = threadIdx.x & 31;
    const int ta = wave / JDIM;                   // 0..63  a-tile
    const int j  = wave % JDIM;                   // 0..63
    const int a0 = ta * 16;
    const int m  = lane & 15;
    const int hi = lane >> 4;

    // Loop-invariant A operand: rows a0..a0+15 of M[:, j, :]
    const float* Abase = M + (a0 + m) * JK + j * KDIM;
    v2f A0 = *(const v2f*)(Abase + 2 * hi);       // K = {0,1} / {2,3}
    v2f A1 = *(const v2f*)(Abase + 4 + 2 * hi);   // K = {4,5} / {6,7}

    float n2a[8];
    #pragma unroll
    for (int r = 0; r < 8; ++r)
        n2a[r] = n2[(a0 + r + 8 * hi) * JDIM + j];

    float fsum[8];
    #pragma unroll
    for (int r = 0; r < 8; ++r) fsum[r] = 0.0f;

    for (int b0 = 0; b0 < BATCH; b0 += 16) {
        const float* Bbase = M + (b0 + m) * JK + j * KDIM;
        v2f B0 = *(const v2f*)(Bbase + 2 * hi);
        v2f B1 = *(const v2f*)(Bbase + 4 + 2 * hi);

        v8f dot = {};
        dot = __builtin_amdgcn_wmma_f32_16x16x4_f32(false, A0, false, B0,
                                                    (short)0, dot, false, false);
        dot = __builtin_amdgcn_wmma_f32_16x16x4_f32(false, A1, false, B1,
                                                    (short)0, dot, false, false);

        const float nb = n2[(b0 + m) * JDIM + j];   // indexed by N = lane&15

        #pragma unroll
        for (int r = 0; r < 8; ++r) {
            float sq  = n2a[r] + nb - 2.0f * dot[r];
            float nrm = (sq > 0.0f) ? __builtin_amdgcn_sqrtf(sq) : 0.0f;
            float t   = fminf(nrm, CLAMP_MAX);
            fsum[r] += __builtin_amdgcn_exp2f(t * NEG_LOG2E);
        }
    }

    // Reduce each accumulator across its 16-lane half; lanes 0 and 16 write.
    #pragma unroll
    for (int r = 0; r < 8; ++r) {
        float v = fsum[r];
        v += __shfl_xor(v, 1, 32);
        v += __shfl_xor(v, 2, 32);
        v += __shfl_xor(v, 4, 32);
        v += __shfl_xor(v, 8, 32);
        if (m == 0)
            out[(a0 + r + 8 * hi) * OUTC + INF + j] = v;
    }
}

// ---------------------------------------------------------------------------
// Kernel 4: out[b, 0:512] = x[b, :]   (float4 copy, 576-col rows stay 16B-aligned)
// ---------------------------------------------------------------------------
__global__ __launch_bounds__(256) void copy_x(const float* __restrict__ x,
                                              float* __restrict__ out) {
    const int idx = blockIdx.x * blockDim.x + threadIdx.x;
    const int n4  = BATCH * INF / 4;
    if (idx < n4) {
        const int row = idx / (INF / 4);
        const int c4  = idx % (INF / 4);
        float4 v = ((const float4*)x)[idx];
        ((float4*)(out + row * OUTC))[c4] = v;
    }
}

extern "C" void kernel_launch(void* const* d_in, const int* in_sizes, int n_in,
                              void* d_out, int out_size, void* d_ws, size_t ws_size,
                              hipStream_t stream) {
    (void)in_sizes; (void)n_in; (void)out_size; (void)ws_size;
    const float* x = (const float*)d_in[0];
    const float* T = (const float*)d_in[1];
    float* out = (float*)d_out;

    float* M  = (float*)d_ws;            // BATCH*JK   = 2 MB
    float* n2 = M + BATCH * JK;          // BATCH*JDIM = 256 KB

    // GEMM1: 64x32 = 2048 waves (one 16x16 tile each) -> 256 blocks of 8 waves
    gemm_m<<<(2048 * 32) / 256, 256, 0, stream>>>(x, T, M);

    // Row norms: one thread per (b,j)
    row_norms<<<(BATCH * JDIM + 255) / 256, 256, 0, stream>>>(M, n2);

    // Copy x into left block of the output (independent of M)
    copy_x<<<(BATCH * INF / 4 + 255) / 256, 256, 0, stream>>>(x, out);

    // Pairwise features: 64 a-tiles x 64 j = 4096 waves -> 512 blocks
    pairwise_feats<<<(4096 * 32) / 256, 256, 0, stream>>>(M, n2, out);
}